// DNN_Beamformer_20486994002318
// MI455X (gfx1250) — compile-verified
//
#include <hip/hip_runtime.h>
#include <hip/hip_bf16.h>

#define B_DIM 8
#define T_DIM 1000
#define C_DIM 8
#define F_DIM 257
#define A_DIM 320

typedef __attribute__((ext_vector_type(2))) float v2f;
typedef __attribute__((ext_vector_type(8))) float v8f;

// ---------------------------------------------------------------------------
// Cooperative PSD kernel: one workgroup (16 waves, 512 threads) per (b, f-tile
// of 16).  Data slabs (8t x 8c x 16f, re+im) are staged Global->LDS with
// CDNA5 async-to-LDS loads (ASYNCcnt), double-buffered and overlapped with the
// V_WMMA_F32_16X16X4_F32 K-loop.  Each wave owns one f and computes the
// complex 8x8 PSD packed as a 16x16 real GEMM:
//   A = [Re(x*m); Im(x*m)] (16 x T), B = [Re(x); Im(x)] (16 x T), D = A*B^T
//   Re(psd[c][e]) = D[c][e] + D[c+8][e+8]
//   Im(psd[c][e]) = D[c+8][e] - D[c][e+8]
// LDS rows padded to 20 floats (16B-aligned rows, spread banks).
// ---------------------------------------------------------------------------

#define LROW 20                      // padded floats per (t,c) row
#define CHUNK_T 8                    // t's staged per chunk
#define SLAB (64 * LROW)             // floats per (buf,comp) slab = 1280

__device__ __forceinline__ void stage_chunk(
    unsigned sbase, int buf, int tchunk, int b, int fbase, int tid,
    const float* __restrict__ dre, const float* __restrict__ dimag) {
#pragma unroll
  for (int p = 0; p < 4; ++p) {
    const int flat = p * 512 + tid;          // 0..2047 elements this chunk
    const int comp = flat >> 10;             // 0: re, 1: im (uniform per p)
    const int rem  = flat & 1023;
    const int row  = rem >> 4;               // t_local*8 + c  (0..63)
    const int fw   = rem & 15;               // f within tile
    const int t_local = row >> 3;
    const int c       = row & 7;
    const float* src  = comp ? dimag : dre;
    const size_t gi = ((size_t)(b * T_DIM + tchunk + t_local) * C_DIM + c) * F_DIM
                    + (size_t)(fbase + fw);
    const unsigned long long ga = (unsigned long long)(const void*)(src + gi);
    const unsigned loff =
        sbase + (((unsigned)((buf * 2 + comp) * SLAB + row * LROW + fw)) << 2);
    asm volatile("global_load_async_to_lds_b32 %0, %1, off"
                 :: "v"(loff), "v"(ga) : "memory");
  }
}

__global__ __launch_bounds__(512) void psd_coop_kernel(
    const float* __restrict__ dre, const float* __restrict__ dimag,
    const float* __restrict__ msk_s, const float* __restrict__ msk_n,
    float* __restrict__ psd_s_re, float* __restrict__ psd_s_im,
    float* __restrict__ psd_n_re, float* __restrict__ psd_n_im) {
  __shared__ float sbuf[2 * 2 * SLAB];       // [buf][comp][64 rows x LROW]
  const int tid  = (int)threadIdx.x;
  const int wid  = tid >> 5;                 // wave id = f within tile
  const int lane = tid & 31;
  const int b     = blockIdx.x >> 4;
  const int ftile = blockIdx.x & 15;
  const int fbase = ftile * 16;              // 16B-aligned f base (f<256)
  const int f     = fbase + wid;

  // LDS byte offset: low 32 bits of the generic pointer ARE the LDS address.
  const unsigned sbase = (unsigned)(unsigned long long)(void*)&sbuf[0];

  // --- per-wave mask sums over T (normalization m = mask/(sum+eps)) --------
  const int c    = lane & 7;
  const int isIm = (lane >> 3) & 1;
  const int toff = (lane >> 4) * 2;
  const size_t mb = ((size_t)(b * F_DIM + f) * C_DIM + c) * T_DIM;
  float ssum = 0.0f, nsum = 0.0f;
  for (int t = (lane >> 3); t < T_DIM; t += 4) {
    ssum += msk_s[mb + t];
    nsum += msk_n[mb + t];
  }
  ssum += __shfl_xor(ssum, 8, 32);  ssum += __shfl_xor(ssum, 16, 32);
  nsum += __shfl_xor(nsum, 8, 32);  nsum += __shfl_xor(nsum, 16, 32);
  const float inv_s = 1.0f / (ssum + 1e-6f);
  const float inv_n = 1.0f / (nsum + 1e-6f);

  // --- software pipeline: async stage chunk k+1 while WMMA-ing chunk k -----
  v8f acc_s = {0.f, 0.f, 0.f, 0.f, 0.f, 0.f, 0.f, 0.f};
  v8f acc_n = {0.f, 0.f, 0.f, 0.f, 0.f, 0.f, 0.f, 0.f};
  const int NCH = T_DIM / CHUNK_T;           // 125
  stage_chunk(sbase, 0, 0, b, fbase, tid, dre, dimag);
  for (int chunk = 0; chunk < NCH; ++chunk) {
    const int buf = chunk & 1;
    if (chunk + 1 < NCH) {
      stage_chunk(sbase, (chunk + 1) & 1, (chunk + 1) * CHUNK_T, b, fbase, tid,
                  dre, dimag);
      asm volatile("s_wait_asynccnt 4" ::: "memory");   // chunk's 4 ops done
    } else {
      asm volatile("s_wait_asynccnt 0" ::: "memory");
    }
    __syncthreads();                                     // all 16 waves landed
    const float* lb = sbuf + (buf * 2 + isIm) * SLAB;
    const int tch = chunk * CHUNK_T;
#pragma unroll
    for (int s = 0; s < 2; ++s) {
      const int tl = s * 4 + toff;           // this lane's K-slice {tl, tl+1}
      const float bx = lb[(tl * 8 + c) * LROW + wid];
      const float by = lb[((tl + 1) * 8 + c) * LROW + wid];
      const int t = tch + tl;
      const float ms0 = msk_s[mb + t] * inv_s, ms1 = msk_s[mb + t + 1] * inv_s;
      const float mn0 = msk_n[mb + t] * inv_n, mn1 = msk_n[mb + t + 1] * inv_n;
      v2f bv;  bv.x = bx;        bv.y = by;
      v2f as_; as_.x = bx * ms0; as_.y = by * ms1;
      v2f an_; an_.x = bx * mn0; an_.y = by * mn1;
      acc_s = __builtin_amdgcn_wmma_f32_16x16x4_f32(
          false, as_, false, bv, (short)0, acc_s, false, false);
      acc_n = __builtin_amdgcn_wmma_f32_16x16x4_f32(
          false, an_, false, bv, (short)0, acc_n, false, false);
    }
    __syncthreads();                         // buffer safe to overwrite
  }

  // --- fold 16x16 D into complex 8x8 and store -----------------------------
  // Acc: VGPR r, lane L<16 -> (M=r, N=L); lane L>=16 -> (M=r+8, N=L-16)
  const size_t obase = ((size_t)(b * F_DIM + f)) * 64;
  int s1, s2;
  if (lane < 8)       { s1 = lane;     s2 = lane + 24; }  // Re
  else if (lane < 16) { s1 = lane + 8; s2 = lane;      }  // Im
  else                { s1 = 0;        s2 = 0;         }
#pragma unroll
  for (int r = 0; r < 8; ++r) {
    const float vs = acc_s[r];
    const float a1 = __shfl(vs, s1, 32), a2 = __shfl(vs, s2, 32);
    const float res_s = (lane < 8) ? (a1 + a2) : (a1 - a2);
    const float vn = acc_n[r];
    const float b1 = __shfl(vn, s1, 32), b2 = __shfl(vn, s2, 32);
    const float res_n = (lane < 8) ? (b1 + b2) : (b1 - b2);
    if (lane < 8) {
      psd_s_re[obase + r * 8 + lane] = res_s;
      psd_n_re[obase + r * 8 + lane] = res_n;
    } else if (lane < 16) {
      psd_s_im[obase + r * 8 + (lane - 8)] = res_s;
      psd_n_im[obase + r * 8 + (lane - 8)] = res_n;
    }
  }
}

// ---------------------------------------------------------------------------
// Edge PSD kernel: f = 256 only (one wave per b), direct-gather variant so the
// async tiles above stay 16-float aligned and fully in-bounds.
// ---------------------------------------------------------------------------
__global__ __launch_bounds__(32) void psd_edge_kernel(
    const float* __restrict__ dre, const float* __restrict__ dimag,
    const float* __restrict__ msk_s, const float* __restrict__ msk_n,
    float* __restrict__ psd_s_re, float* __restrict__ psd_s_im,
    float* __restrict__ psd_n_re, float* __restrict__ psd_n_im) {
  const int b = blockIdx.x;
  const int f = F_DIM - 1;
  const int lane = (int)threadIdx.x;
  const int c = lane & 7;
  const int isIm = (lane >> 3) & 1;
  const int toff = (lane >> 4) * 2;

  const size_t mb = ((size_t)(b * F_DIM + f) * C_DIM + c) * T_DIM;
  float ss = 0.0f, sn = 0.0f;
  for (int t = (lane >> 3); t < T_DIM; t += 4) {
    ss += msk_s[mb + t];
    sn += msk_n[mb + t];
  }
  ss += __shfl_xor(ss, 8, 32);  ss += __shfl_xor(ss, 16, 32);
  sn += __shfl_xor(sn, 8, 32);  sn += __shfl_xor(sn, 16, 32);
  const float inv_s = 1.0f / (ss + 1e-6f);
  const float inv_n = 1.0f / (sn + 1e-6f);

  v8f acc_s = {0.f, 0.f, 0.f, 0.f, 0.f, 0.f, 0.f, 0.f};
  v8f acc_n = {0.f, 0.f, 0.f, 0.f, 0.f, 0.f, 0.f, 0.f};
  const size_t dstride = (size_t)C_DIM * F_DIM;
  for (int t0 = 0; t0 < T_DIM; t0 += 4) {
    const int t = t0 + toff;
    const size_t di = ((size_t)(b * T_DIM + t) * C_DIM + c) * F_DIM + f;
    const float xr0 = dre[di],           xi0 = dimag[di];
    const float xr1 = dre[di + dstride], xi1 = dimag[di + dstride];
    const float ms0 = msk_s[mb + t]     * inv_s;
    const float ms1 = msk_s[mb + t + 1] * inv_s;
    const float mn0 = msk_n[mb + t]     * inv_n;
    const float mn1 = msk_n[mb + t + 1] * inv_n;
    v2f bv, as_, an_;
    bv.x = isIm ? xi0 : xr0;
    bv.y = isIm ? xi1 : xr1;
    as_.x = bv.x * ms0;  as_.y = bv.y * ms1;
    an_.x = bv.x * mn0;  an_.y = bv.y * mn1;
    acc_s = __builtin_amdgcn_wmma_f32_16x16x4_f32(
        false, as_, false, bv, (short)0, acc_s, false, false);
    acc_n = __builtin_amdgcn_wmma_f32_16x16x4_f32(
        false, an_, false, bv, (short)0, acc_n, false, false);
  }

  const size_t obase = ((size_t)(b * F_DIM + f)) * 64;
  int s1, s2;
  if (lane < 8)       { s1 = lane;     s2 = lane + 24; }
  else if (lane < 16) { s1 = lane + 8; s2 = lane;      }
  else                { s1 = 0;        s2 = 0;         }
#pragma unroll
  for (int r = 0; r < 8; ++r) {
    const float vs = acc_s[r];
    const float a1 = __shfl(vs, s1, 32), a2 = __shfl(vs, s2, 32);
    const float res_s = (lane < 8) ? (a1 + a2) : (a1 - a2);
    const float vn = acc_n[r];
    const float b1 = __shfl(vn, s1, 32), b2 = __shfl(vn, s2, 32);
    const float res_n = (lane < 8) ? (b1 + b2) : (b1 - b2);
    if (lane < 8) {
      psd_s_re[obase + r * 8 + lane] = res_s;
      psd_n_re[obase + r * 8 + lane] = res_n;
    } else if (lane < 16) {
      psd_s_im[obase + r * 8 + (lane - 8)] = res_s;
      psd_n_im[obase + r * 8 + (lane - 8)] = res_n;
    }
  }
}

// ---------------------------------------------------------------------------
// Kernel 2: attention logits e[b][c] = w_gvec . tanh(feat @ w_mlp^T + b_mlp)
// ---------------------------------------------------------------------------
__global__ __launch_bounds__(256) void attention_kernel(
    const float* __restrict__ psd_s_re, const float* __restrict__ psd_s_im,
    const float* __restrict__ w_mlp, const float* __restrict__ b_mlp,
    const float* __restrict__ w_gvec, const float* __restrict__ b_gvec,
    float* __restrict__ e_att) {
  const int b = blockIdx.x >> 3;
  const int c = blockIdx.x & 7;
  const int tid = (int)threadIdx.x;
  __shared__ float feat[F_DIM];
  __shared__ float red[256];

  for (int f = tid; f < F_DIM; f += 256) {
    const size_t base = ((size_t)(b * F_DIM + f) * 8 + c) * 8;
    float sr = 0.0f, si = 0.0f;
    for (int e = 0; e < 8; ++e) { sr += psd_s_re[base + e]; si += psd_s_im[base + e]; }
    sr -= psd_s_re[base + c];
    si -= psd_s_im[base + c];
    sr *= (1.0f / 7.0f);
    si *= (1.0f / 7.0f);
    feat[f] = sqrtf(sr * sr + si * si);
  }
  __syncthreads();

  float acc = 0.0f;
  for (int a = tid; a < A_DIM; a += 256) {
    float h = b_mlp[a];
    const float* wrow = w_mlp + (size_t)a * F_DIM;
    for (int f = 0; f < F_DIM; ++f) h += feat[f] * wrow[f];
    acc += w_gvec[a] * tanhf(h);
  }
  red[tid] = acc;
  __syncthreads();
  for (int s = 128; s > 0; s >>= 1) {
    if (tid < s) red[tid] += red[tid + s];
    __syncthreads();
  }
  if (tid == 0) e_att[blockIdx.x] = red[0] + b_gvec[0];
}

// ---------------------------------------------------------------------------
// Kernel 3: u[b] = softmax(2 * e[b]) over C=8.
// ---------------------------------------------------------------------------
__global__ __launch_bounds__(32) void softmax_kernel(
    const float* __restrict__ e_att, float* __restrict__ u_vec) {
  const int b = blockIdx.x;
  const int lane = (int)threadIdx.x;
  float v = e_att[b * 8 + (lane & 7)] * 2.0f;
  float m = v;
  m = fmaxf(m, __shfl_xor(m, 1, 32));
  m = fmaxf(m, __shfl_xor(m, 2, 32));
  m = fmaxf(m, __shfl_xor(m, 4, 32));
  const float ex = expf(v - m);
  float s = ex;
  s += __shfl_xor(s, 1, 32);
  s += __shfl_xor(s, 2, 32);
  s += __shfl_xor(s, 4, 32);
  if (lane < 8) u_vec[b * 8 + lane] = ex / s;
}

// ---------------------------------------------------------------------------
// Kernel 4: per-(b,f) complex 8x8 Gauss-Jordan (LDS, one wave per tile):
// X = (psd_n + reg*I)^{-1} psd_s; ws[e] = (sum_c X[e][c] u[c]) / (tr(X)+eps).
// ---------------------------------------------------------------------------
__global__ __launch_bounds__(32) void solve_kernel(
    const float* __restrict__ psd_s_re, const float* __restrict__ psd_s_im,
    const float* __restrict__ psd_n_re, const float* __restrict__ psd_n_im,
    const float* __restrict__ u_vec,
    float* __restrict__ ws_re, float* __restrict__ ws_im) {
  const int bf = blockIdx.x;
  const int b = bf / F_DIM, f = bf % F_DIM;
  const int lane = (int)threadIdx.x;
  __shared__ float Ar[8][8], Ai[8][8], Xr[8][8], Xi[8][8];
  __shared__ float fr_[8], fi_[8];
  const size_t base = (size_t)bf * 64;

  for (int q = 0; q < 2; ++q) {
    const int idx = lane + q * 32;
    const int i = idx >> 3, j = idx & 7;
    Ar[i][j] = psd_n_re[base + idx];
    Ai[i][j] = psd_n_im[base + idx];
    Xr[i][j] = psd_s_re[base + idx];
    Xi[i][j] = psd_s_im[base + idx];
  }
  __syncthreads();
  if (lane == 0) {
    float tr = 0.0f;
    for (int i = 0; i < 8; ++i) tr += Ar[i][i];
    const float reg = 1e-7f * (tr * 0.125f) + 1e-8f;
    for (int i = 0; i < 8; ++i) Ar[i][i] += reg;
  }
  __syncthreads();

  for (int k = 0; k < 8; ++k) {
    const float pr = Ar[k][k], pi = Ai[k][k];
    const float dinv = 1.0f / (pr * pr + pi * pi);
    const float ipr = pr * dinv, ipi = -pi * dinv;
    if (lane < 8) {
      const float ar = Ar[k][lane], ai = Ai[k][lane];
      Ar[k][lane] = ar * ipr - ai * ipi;
      Ai[k][lane] = ar * ipi + ai * ipr;
    } else if (lane < 16) {
      const int j = lane - 8;
      const float xr = Xr[k][j], xi = Xi[k][j];
      Xr[k][j] = xr * ipr - xi * ipi;
      Xi[k][j] = xr * ipi + xi * ipr;
    }
    __syncthreads();
    if (lane < 8) { fr_[lane] = Ar[lane][k]; fi_[lane] = Ai[lane][k]; }
    __syncthreads();
    for (int q = 0; q < 4; ++q) {
      const int idx = lane * 4 + q;
      const int i = idx >> 4, j = idx & 15;
      if (i == k) continue;
      const float fr = fr_[i], fi = fi_[i];
      if (j < 8) {
        const float kr = Ar[k][j], ki = Ai[k][j];
        Ar[i][j] -= fr * kr - fi * ki;
        Ai[i][j] -= fr * ki + fi * kr;
      } else {
        const int jj = j - 8;
        const float kr = Xr[k][jj], ki = Xi[k][jj];
        Xr[i][jj] -= fr * kr - fi * ki;
        Xi[i][jj] -= fr * ki + fi * kr;
      }
    }
    __syncthreads();
  }

  float trr = 1e-6f, tri = 0.0f;
  for (int i = 0; i < 8; ++i) { trr += Xr[i][i]; tri += Xi[i][i]; }
  const float dd = 1.0f / (trr * trr + tri * tri);
  if (lane < 8) {
    const int e = lane;
    float sr = 0.0f, si = 0.0f;
    for (int c2 = 0; c2 < 8; ++c2) {
      const float uc = u_vec[b * 8 + c2];
      sr += Xr[e][c2] * uc;
      si += Xi[e][c2] * uc;
    }
    const float wr = (sr * trr + si * tri) * dd;
    const float wi = (si * trr - sr * tri) * dd;
    ws_re[((size_t)b * 8 + e) * F_DIM + f] = wr;
    ws_im[((size_t)b * 8 + e) * F_DIM + f] = wi;
  }
}

// ---------------------------------------------------------------------------
// Kernel 5: enhanced[b][t][f] = sum_c conj(ws[b][f][c]) * x[b][f][c][t]
// ---------------------------------------------------------------------------
__global__ __launch_bounds__(256) void beamform_kernel(
    const float* __restrict__ dre, const float* __restrict__ dimag,
    const float* __restrict__ ws_re, const float* __restrict__ ws_im,
    float* __restrict__ out) {
  const int bt = blockIdx.x;
  const int b = bt / T_DIM, t = bt % T_DIM;
  for (int f = (int)threadIdx.x; f < F_DIM; f += 256) {
    float er = 0.0f, ei = 0.0f;
    size_t di = ((size_t)(b * T_DIM + t) * C_DIM) * F_DIM + f;
    size_t wix = (size_t)b * C_DIM * F_DIM + f;
    for (int c = 0; c < C_DIM; ++c) {
      const float xr = dre[di], xi = dimag[di];
      const float wr = ws_re[wix], wim = ws_im[wix];
      er += wr * xr + wim * xi;
      ei += wr * xi - wim * xr;
      di += F_DIM;
      wix += F_DIM;
    }
    const size_t oi = ((size_t)(b * T_DIM + t) * F_DIM + f) * 2;
    out[oi]     = er;
    out[oi + 1] = ei;
  }
}

extern "C" void kernel_launch(void* const* d_in, const int* in_sizes, int n_in,
                              void* d_out, int out_size, void* d_ws, size_t ws_size,
                              hipStream_t stream) {
  const float* dre    = (const float*)d_in[0];
  const float* dimag  = (const float*)d_in[1];
  const float* msk_s  = (const float*)d_in[2];
  const float* msk_n  = (const float*)d_in[3];
  const float* w_mlp  = (const float*)d_in[4];
  const float* b_mlp  = (const float*)d_in[5];
  const float* w_gvec = (const float*)d_in[6];
  const float* b_gvec = (const float*)d_in[7];
  float* out = (float*)d_out;
  float* ws  = (float*)d_ws;

  const size_t PSD = (size_t)B_DIM * F_DIM * 64;
  float* psd_s_re = ws;
  float* psd_s_im = ws + PSD;
  float* psd_n_re = ws + 2 * PSD;
  float* psd_n_im = ws + 3 * PSD;
  float* e_att    = ws + 4 * PSD;
  float* u_vec    = e_att + 64;
  float* wre      = u_vec + 64;
  float* wim      = wre + (size_t)B_DIM * C_DIM * F_DIM;

  psd_coop_kernel<<<B_DIM * 16, 512, 0, stream>>>(
      dre, dimag, msk_s, msk_n, psd_s_re, psd_s_im, psd_n_re, psd_n_im);
  psd_edge_kernel<<<B_DIM, 32, 0, stream>>>(
      dre, dimag, msk_s, msk_n, psd_s_re, psd_s_im, psd_n_re, psd_n_im);
  attention_kernel<<<B_DIM * C_DIM, 256, 0, stream>>>(
      psd_s_re, psd_s_im, w_mlp, b_mlp, w_gvec, b_gvec, e_att);
  softmax_kernel<<<B_DIM, 32, 0, stream>>>(e_att, u_vec);
  solve_kernel<<<B_DIM * F_DIM, 32, 0, stream>>>(
      psd_s_re, psd_s_im, psd_n_re, psd_n_im, u_vec, wre, wim);
  beamform_kernel<<<B_DIM * T_DIM, 256, 0, stream>>>(dre, dimag, wre, wim, out);
}